// LI2Model_22419729285231
// MI455X (gfx1250) — compile-verified
//
#include <hip/hip_runtime.h>

typedef __attribute__((ext_vector_type(2))) float v2f;
typedef __attribute__((ext_vector_type(8))) float v8f;

#define T_STEPS 300
#define BATCH   1024
#define OUTN    10

// One wave32 per block; each wave owns a 16-batch tile and runs the whole
// 300-step recurrence with all state in LDS. FC layers -> v_wmma_f32_16x16x4_f32.
__global__ __launch_bounds__(32) void snn_lif_wmma_kernel(
    const float* __restrict__ xg,      // [1024,1,10,10]
    const float* __restrict__ conv_w,  // [2,1,4,4]
    const float* __restrict__ conv_b,  // [2]
    const float* __restrict__ fc_w,    // [10,72]
    const float* __restrict__ fc_b,    // [10]
    float* __restrict__ out)           // [spk_rec(300*1024*10) | spk_out(1024*10) | mem_rec(300*1024*10)]
{
    __shared__ float x_s   [16*100];
    __shared__ float encv_s[16*100];
    __shared__ float zenc_s[16*100];
    __shared__ float lifv_s[16*98];
    __shared__ float lifi_s[16*98];
    __shared__ float zlif_s[16*98];
    __shared__ float li1v_s[16*72];
    __shared__ float li1i_s[16*72];
    __shared__ float li2v_s[16*72];
    __shared__ float li2i_s[16*72];

    const int tid  = threadIdx.x;      // 0..31
    const int m    = tid & 15;         // batch row within tile (== WMMA M/N lane)
    const int h    = tid >> 4;         // half: channel for conv, K-half for WMMA
    const int base = blockIdx.x * 16;  // tile batch base

    // ---- init state, stage x into LDS ----
    for (int j = 0; j < 50; ++j) {
        int e = h*50 + j;
        x_s   [m*100 + e] = xg[(base+m)*100 + e];
        encv_s[m*100 + e] = 0.f;
    }
    for (int j = 0; j < 49; ++j) {
        int idx = m*98 + h*49 + j;
        lifv_s[idx] = 0.f; lifi_s[idx] = 0.f;
    }
    for (int j = 0; j < 36; ++j) {
        int idx = m*72 + h*36 + j;
        li1v_s[idx] = 0.f; li1i_s[idx] = 0.f;
        li2v_s[idx] = 0.f; li2i_s[idx] = 0.f;
    }

    // conv weights for this thread's channel (loop-invariant, in VGPRs)
    float wreg[16];
    #pragma unroll
    for (int k = 0; k < 16; ++k) wreg[k] = conv_w[h*16 + k];
    const float cbias = conv_b[h];

    // WMMA B fragment: fc_w^T as 72x16 (N padded 10->16 with zeros).
    // Lane layout: n = tid&15; lanes 0-15 hold K = 4kc,4kc+1; lanes 16-31 K = 4kc+2,4kc+3.
    const float bias = (m < OUTN) ? fc_b[m] : 0.f;
    v2f bfrag[18];
    #pragma unroll
    for (int kc = 0; kc < 18; ++kc) {
        int k0 = kc*4 + h*2;
        float b0 = 0.f, b1 = 0.f;
        if (m < OUTN) { b0 = fc_w[m*72 + k0]; b1 = fc_w[m*72 + k0 + 1]; }
        bfrag[kc].x = b0; bfrag[kc].y = b1;
    }

    v8f spk_acc;
    #pragma unroll
    for (int r = 0; r < 8; ++r) spk_acc[r] = 0.f;

    __syncthreads();

    #pragma unroll 1
    for (int t = 0; t < T_STEPS; ++t) {
        // --- constant-current LIF encode: v = 0.9v + 0.1x; z = (v>1); v *= (1-z) ---
        #pragma unroll 1
        for (int j = 0; j < 50; ++j) {
            int a = m*100 + h*50 + j;
            float v = 0.9f*encv_s[a] + 0.1f*x_s[a];
            float z = (v > 1.0f) ? 1.0f : 0.0f;
            encv_s[a] = (1.0f - z) * v;
            zenc_s[a] = z;
        }
        __syncthreads();

        // --- conv(10x10 -> 7x7, ch = h) + LIF cell ---
        #pragma unroll 1
        for (int r = 0; r < 7; ++r) {
            #pragma unroll 1
            for (int c = 0; c < 7; ++c) {
                float cur = cbias;
                #pragma unroll
                for (int i = 0; i < 4; ++i)
                    #pragma unroll
                    for (int j = 0; j < 4; ++j)
                        cur += wreg[i*4+j] * zenc_s[m*100 + (r+i)*10 + (c+j)];
                int idx = m*98 + h*49 + r*7 + c;
                float v_old = lifv_s[idx], i_old = lifi_s[idx];
                float v_dec = 0.9f*v_old + 0.1f*i_old;   // v + dt/tau_m*(-v + i)
                float i_dec = 0.8f*i_old;                // i * (1 - dt/tau_s)
                float z = (v_dec > 1.0f) ? 1.0f : 0.0f;
                lifv_s[idx] = (1.0f - z) * v_dec;
                lifi_s[idx] = i_dec + cur;
                zlif_s[idx] = z;
            }
        }
        __syncthreads();

        // --- 2x2/s1 maxpool (7x7 -> 6x6) + two LI cells ---
        #pragma unroll 1
        for (int rr = 0; rr < 6; ++rr) {
            #pragma unroll 1
            for (int cc = 0; cc < 6; ++cc) {
                int p0 = m*98 + h*49 + rr*7 + cc;
                float pm = fmaxf(fmaxf(lifv_s[p0], lifv_s[p0+1]),
                                 fmaxf(lifv_s[p0+7], lifv_s[p0+8]));
                float ps = fmaxf(fmaxf(zlif_s[p0], zlif_s[p0+1]),
                                 fmaxf(zlif_s[p0+7], zlif_s[p0+8]));
                int li = m*72 + h*36 + rr*6 + cc;
                float v1 = li1v_s[li], i1 = li1i_s[li];
                li1v_s[li] = 0.9f*v1 + 0.1f*i1;
                li1i_s[li] = 0.8f*i1 + pm;
                float v2 = li2v_s[li], i2 = li2i_s[li];
                li2v_s[li] = 0.9f*v2 + 0.1f*i2;
                li2i_s[li] = 0.8f*i2 + ps;
            }
        }
        __syncthreads();

        // --- FC layers: D = A[16x72] * fc_w^T[72x16] + bias, via f32 WMMA, K = 18 chunks of 4 ---
        v8f cm, cs;
        #pragma unroll
        for (int r = 0; r < 8; ++r) { cm[r] = bias; cs[r] = bias; }

        #pragma unroll
        for (int kc = 0; kc < 18; ++kc) {
            int k0 = kc*4 + h*2;
            v2f a1, a2;
            a1.x = li1v_s[m*72 + k0]; a1.y = li1v_s[m*72 + k0 + 1];
            a2.x = li2v_s[m*72 + k0]; a2.y = li2v_s[m*72 + k0 + 1];
            cm = __builtin_amdgcn_wmma_f32_16x16x4_f32(
                false, a1, false, bfrag[kc], (short)0, cm, false, false);
            cs = __builtin_amdgcn_wmma_f32_16x16x4_f32(
                false, a2, false, bfrag[kc], (short)0, cs, false, false);
        }

        // spk_out accumulator (sum over time of out_spk)
        #pragma unroll
        for (int r = 0; r < 8; ++r) spk_acc[r] += cs[r];

        // write spk_rec (from li2 path) and mem_rec (from li1 path)
        if (m < OUTN) {
            #pragma unroll
            for (int r = 0; r < 8; ++r) {
                int  row = r + 8*h;                    // C/D layout: M = r + 8*(lane>=16)
                long gb  = base + row;
                out[(long)t*BATCH*OUTN + gb*OUTN + m] = cs[r];
                out[(long)T_STEPS*BATCH*OUTN + (long)BATCH*OUTN
                    + (long)t*BATCH*OUTN + gb*OUTN + m] = cm[r];
            }
        }
        __syncthreads();
    }

    // spk_out
    if (m < OUTN) {
        #pragma unroll
        for (int r = 0; r < 8; ++r) {
            int row = r + 8*h;
            out[(long)T_STEPS*BATCH*OUTN + (long)(base+row)*OUTN + m] = spk_acc[r];
        }
    }
}

extern "C" void kernel_launch(void* const* d_in, const int* in_sizes, int n_in,
                              void* d_out, int out_size, void* d_ws, size_t ws_size,
                              hipStream_t stream) {
    const float* x      = (const float*)d_in[0];  // [1024,1,10,10]
    const float* conv_w = (const float*)d_in[1];  // [2,1,4,4]
    const float* conv_b = (const float*)d_in[2];  // [2]
    const float* fc_w   = (const float*)d_in[3];  // [10,72]
    const float* fc_b   = (const float*)d_in[4];  // [10]
    float* out = (float*)d_out;

    dim3 grid(BATCH / 16);   // 64 tiles of 16 batch, one wave32 each
    dim3 block(32);
    snn_lif_wmma_kernel<<<grid, block, 0, stream>>>(x, conv_w, conv_b, fc_w, fc_b, out);
}